// FDiff_7885559956093
// MI455X (gfx1250) — compile-verified
//
#include <hip/hip_runtime.h>
#include <hip/hip_bf16.h>
#include <math.h>

#define N_NODES 100000
#define FEATS   512
#define HID     64
#define CLS     64
#define NE      3200000
#define DEPTH   10

typedef __attribute__((ext_vector_type(2))) float v2f;
typedef __attribute__((ext_vector_type(8))) float v8f;

// ---------------------------------------------------------------------------
// Degree accumulation: one thread per edge, int atomics.
// ---------------------------------------------------------------------------
__global__ void k_degree(const int* __restrict__ src, const int* __restrict__ dst,
                         int* __restrict__ degs, int* __restrict__ degd) {
    int e = blockIdx.x * blockDim.x + threadIdx.x;
    if (e >= NE) return;
    atomicAdd(&degs[src[e]], 1);
    atomicAdd(&degd[dst[e]], 1);
}

// norm[e] = rsqrt(max(deg_out[src],1) * max(deg_in[dst],1))
__global__ void k_norm(const int* __restrict__ src, const int* __restrict__ dst,
                       const int* __restrict__ degs, const int* __restrict__ degd,
                       float* __restrict__ norm) {
    int e = blockIdx.x * blockDim.x + threadIdx.x;
    if (e >= NE) return;
    float a = (float)max(degs[src[e]], 1);
    float b = (float)max(degd[dst[e]], 1);
    norm[e] = rsqrtf(a * b);
}

// ---------------------------------------------------------------------------
// f32 WMMA GEMM: out[M x 64] = A[M x K] @ B[K x 64] (+bias, optional relu).
// One wave computes a 16x16 output tile using V_WMMA_F32_16X16X4_F32.
// A frag (16x4):  lane m=lane&15 ; kb=(lane>>4)*2 ; a = {A[m][k+kb], A[m][k+kb+1]}
// B frag (4x16):  lane n=lane&15 ; b = {B[k+kb][n], B[k+kb+1][n]}
// D (16x16 f32):  VGPR v -> row v + 8*(lane>>4), col = lane&15
// ---------------------------------------------------------------------------
__global__ void k_wmma_gemm(const float* __restrict__ A, const float* __restrict__ B,
                            const float* __restrict__ bias, float* __restrict__ out,
                            int K, int relu) {
    int lane  = threadIdx.x & 31;
    int gwave = blockIdx.x * (blockDim.x >> 5) + (threadIdx.x >> 5);
    int rowTile = gwave >> 2;           // 64 cols -> 4 col tiles
    int colTile = gwave & 3;
    if (rowTile >= N_NODES / 16) return;   // uniform per wave: EXEC stays all-ones

    int m  = lane & 15;
    int kb = (lane >> 4) * 2;
    int n0 = colTile * 16;
    const float* arow = A + (size_t)(rowTile * 16 + m) * K;

    v8f acc = {};
    for (int k = 0; k < K; k += 4) {
        v2f a, b;
        a.x = arow[k + kb];
        a.y = arow[k + kb + 1];
        b.x = B[(size_t)(k + kb)     * 64 + n0 + m];
        b.y = B[(size_t)(k + kb + 1) * 64 + n0 + m];
        acc = __builtin_amdgcn_wmma_f32_16x16x4_f32(false, a, false, b,
                                                    (short)0, acc, false, false);
    }

    int   col = n0 + m;
    float bv  = bias[col];
    int   rbase = rowTile * 16 + (lane >> 4) * 8;
#pragma unroll
    for (int v = 0; v < 8; ++v) {
        float val = acc[v] + bv;
        if (relu) val = val > 0.0f ? val : 0.0f;
        out[(size_t)(rbase + v) * 64 + col] = val;
    }
}

// ---------------------------------------------------------------------------
// Wave-per-node softmax (64 classes, 2 per lane, wave32 shfl reductions),
// then err = onehot(labels)*train - p. Writes p in place, err to two buffers.
// ---------------------------------------------------------------------------
__global__ void k_softmax_err(float* __restrict__ p, const int* __restrict__ mask,
                              const int* __restrict__ labels,
                              float* __restrict__ errA, float* __restrict__ errH0) {
    int lane = threadIdx.x & 31;
    int node = (blockIdx.x * blockDim.x + threadIdx.x) >> 5;
    if (node >= N_NODES) return;
    float* row = p + (size_t)node * 64;
    float v0 = row[lane], v1 = row[lane + 32];
    float mx = fmaxf(v0, v1);
#pragma unroll
    for (int off = 16; off > 0; off >>= 1) mx = fmaxf(mx, __shfl_xor(mx, off, 32));
    float e0 = expf(v0 - mx), e1 = expf(v1 - mx);
    float s = e0 + e1;
#pragma unroll
    for (int off = 16; off > 0; off >>= 1) s += __shfl_xor(s, off, 32);
    float inv = 1.0f / s;
    float p0 = e0 * inv, p1 = e1 * inv;
    int train = (mask[node] != 0);
    int lab = labels[node];
    float o0 = (train && lab == lane)      ? 1.0f : 0.0f;
    float o1 = (train && lab == lane + 32) ? 1.0f : 0.0f;
    row[lane] = p0; row[lane + 32] = p1;
    size_t base = (size_t)node * 64;
    float d0 = o0 - p0, d1 = o1 - p1;
    errA[base + lane] = d0;  errA[base + lane + 32] = d1;
    errH0[base + lane] = d0; errH0[base + lane + 32] = d1;
}

// ---------------------------------------------------------------------------
// Graph conv scatter: one wave per edge. Coalesced 256B gather of row src[e],
// scaled by norm[e], contiguous float atomic scatter into row dst[e].
// ---------------------------------------------------------------------------
__global__ void k_scatter(const float* __restrict__ in, float* __restrict__ out,
                          const int* __restrict__ src, const int* __restrict__ dst,
                          const float* __restrict__ norm) {
    long long tid = (long long)blockIdx.x * blockDim.x + threadIdx.x;
    int e    = (int)(tid >> 5);
    int lane = (int)(tid & 31);
    if (e >= NE) return;
    int s = src[e], d = dst[e];
    float nm = norm[e];
    const float2* ip = (const float2*)(in + (size_t)s * 64) + lane;
    float2 v = *ip;
    float* op = out + (size_t)d * 64 + lane * 2;
    atomicAdd(op,     v.x * nm);
    atomicAdd(op + 1, v.y * nm);
}

// err = where(train, h0, conv_result)
__global__ void k_select_train(const float* __restrict__ bsrc, const float* __restrict__ h0,
                               const int* __restrict__ mask, float* __restrict__ aout) {
    int i = blockIdx.x * blockDim.x + threadIdx.x;
    if (i >= N_NODES * 64) return;
    aout[i] = (mask[i >> 6] != 0) ? h0[i] : bsrc[i];
}

// out0 = p + err ; store to both current buffer and h0 of loop 2
__global__ void k_init_out(const float* __restrict__ p, float* __restrict__ aio,
                           float* __restrict__ h0out) {
    int i = blockIdx.x * blockDim.x + threadIdx.x;
    if (i >= N_NODES * 64) return;
    float v = p[i] + aio[i];
    aio[i] = v;
    h0out[i] = v;
}

// out = conv*0.9 + h0*0.1
__global__ void k_blend(const float* __restrict__ bsrc, const float* __restrict__ h0,
                        float* __restrict__ aout) {
    int i = blockIdx.x * blockDim.x + threadIdx.x;
    if (i >= N_NODES * 64) return;
    aout[i] = bsrc[i] * 0.9f + h0[i] * 0.1f;
}

__global__ void k_final_log(const float* __restrict__ a, float* __restrict__ out) {
    int i = blockIdx.x * blockDim.x + threadIdx.x;
    if (i >= N_NODES * 64) return;
    out[i] = logf(a[i] + 1.0f);
}

// ---------------------------------------------------------------------------
extern "C" void kernel_launch(void* const* d_in, const int* in_sizes, int n_in,
                              void* d_out, int out_size, void* d_ws, size_t ws_size,
                              hipStream_t stream) {
    const float* x      = (const float*)d_in[0];
    const float* w1     = (const float*)d_in[1];
    const float* b1     = (const float*)d_in[2];
    const float* w2     = (const float*)d_in[3];
    const float* b2     = (const float*)d_in[4];
    const int*   edges  = (const int*)d_in[5];
    const int*   mask   = (const int*)d_in[6];
    const int*   labels = (const int*)d_in[7];
    const int*   src    = edges;
    const int*   dst    = edges + NE;
    float* out = (float*)d_out;

    const size_t NF = (size_t)N_NODES * 64;
    float* norm  = (float*)d_ws;          // NE floats
    float* h     = norm  + NE;            // NF  (hidden; reused as h0 of loop 2)
    float* p     = h     + NF;            // NF  (logits -> probabilities)
    float* bufA  = p     + NF;            // NF  (current err / out)
    float* bufB  = bufA  + NF;            // NF  (conv accumulator)
    float* errH0 = bufB  + NF;            // NF  (pinned err for loop 1)
    int*   degs  = (int*)(errH0 + NF);    // N
    int*   degd  = degs + N_NODES;        // N

    const int B = 256;
    dim3 blk(B);
    dim3 gEdge((NE + B - 1) / B);                       // 12500
    dim3 gGemm((N_NODES / 16) * 4 / (B / 32));          // 3125 (exact)
    dim3 gRow((N_NODES * 32 + B - 1) / B);              // 12500
    dim3 gScat((int)(((long long)NE * 32 + B - 1) / B));// 400000
    dim3 gElem((int)((NF + B - 1) / B));                // 25000

    // degrees + edge norms
    hipMemsetAsync(degs, 0, 2 * (size_t)N_NODES * sizeof(int), stream);
    k_degree<<<gEdge, blk, 0, stream>>>(src, dst, degs, degd);
    k_norm<<<gEdge, blk, 0, stream>>>(src, dst, degs, degd, norm);

    // h = relu(x@w1+b1) ; logits = h@w2+b2  (f32 WMMA)
    k_wmma_gemm<<<gGemm, blk, 0, stream>>>(x, w1, b1, h, FEATS, 1);
    k_wmma_gemm<<<gGemm, blk, 0, stream>>>(h, w2, b2, p, HID, 0);

    // p = softmax(logits), err = onehot*train - p
    k_softmax_err<<<gRow, blk, 0, stream>>>(p, mask, labels, bufA, errH0);

    // loop 1: label propagation of the error
    for (int d = 0; d < DEPTH; ++d) {
        hipMemsetAsync(bufB, 0, NF * sizeof(float), stream);
        k_scatter<<<gScat, blk, 0, stream>>>(bufA, bufB, src, dst, norm);
        k_select_train<<<gElem, blk, 0, stream>>>(bufB, errH0, mask, bufA);
    }

    // out0 = p + err  (h buffer becomes h0 of loop 2)
    k_init_out<<<gElem, blk, 0, stream>>>(p, bufA, h);

    // loop 2: smoothing
    for (int d = 0; d < DEPTH; ++d) {
        hipMemsetAsync(bufB, 0, NF * sizeof(float), stream);
        k_scatter<<<gScat, blk, 0, stream>>>(bufA, bufB, src, dst, norm);
        k_blend<<<gElem, blk, 0, stream>>>(bufB, h, bufA);
    }

    k_final_log<<<gElem, blk, 0, stream>>>(bufA, out);
}